// Mamba_22539988369531
// MI455X (gfx1250) — compile-verified
//
#include <hip/hip_runtime.h>
#include <math.h>

typedef float v2f __attribute__((ext_vector_type(2)));
typedef float v8f __attribute__((ext_vector_type(8)));

#define T_SEQ 1024
#define H_DIM 128
#define E_DIM 256
#define TCH   8     // timesteps per async-staged chunk in the scan

__device__ __forceinline__ float silu_f(float x) {
    return x / (1.0f + __expf(-x));
}
__device__ __forceinline__ float softplus_f(float x) {
    // jax.nn.softplus = max(x,0) + log1p(exp(-|x|))
    return fmaxf(x, 0.0f) + log1pf(__expf(-fabsf(x)));
}

// ---------------- rmsnorm: one block per row (t), 128 threads ----------------
__global__ void rmsnorm_kernel(const float* __restrict__ x,
                               const float* __restrict__ w,
                               float* __restrict__ out) {
    __shared__ float red[H_DIM];
    int t = blockIdx.x, h = threadIdx.x;
    float v = x[t * H_DIM + h];
    red[h] = v * v;
    __syncthreads();
    for (int s = 64; s > 0; s >>= 1) {
        if (h < s) red[h] += red[h + s];
        __syncthreads();
    }
    float r = rsqrtf(red[0] * (1.0f / H_DIM) + 1e-5f);
    out[t * H_DIM + h] = v * r * w[h];
}

// ---------------- fp32 WMMA GEMM: C[M,N] = A[M,K] * Bw[N,K]^T (+bias +resid) -
// one wave per 16x16 tile, 8 waves (256 threads) per block.
// v_wmma_f32_16x16x4_f32 fragment layout (wave32):
//   A 16x4:  lane&15 -> M row, lane>>4 -> K half; {v0,v1} = A[m, kb], A[m, kb+1]
//   B 4x16:  lane&15 -> N col, lane>>4 -> K half; {v0,v1} = Bw[n, kb], Bw[n, kb+1]
//   D 16x16: vgpr r, lanes 0-15: (M=r, N=lane); lanes 16-31: (M=r+8, N=lane-16)
__global__ void wmma_gemm_f32(const float* __restrict__ A, int lda,
                              const float* __restrict__ Bw, int ldb,
                              const float* __restrict__ bias,
                              const float* __restrict__ resid,
                              float* __restrict__ C, int ldc,
                              int tiles_n, int K) {
    int lane = threadIdx.x & 31;
    int wave = threadIdx.x >> 5;
    int tile = blockIdx.x * 8 + wave;
    int tm = tile / tiles_n;
    int tn = tile % tiles_n;

    int ksub = (lane >> 4) * 2;
    const float* ap = A  + (tm * 16 + (lane & 15)) * (size_t)lda + ksub;
    const float* bp = Bw + (tn * 16 + (lane & 15)) * (size_t)ldb + ksub;

    v8f acc = {};
    for (int k = 0; k < K; k += 4) {
        v2f a = *(const v2f*)(ap + k);
        v2f b = *(const v2f*)(bp + k);
        acc = __builtin_amdgcn_wmma_f32_16x16x4_f32(
            false, a, false, b, (short)0, acc, false, false);
    }

    int col   = tn * 16 + (lane & 15);
    int rbase = tm * 16 + ((lane >> 4) * 8);
    float bv = bias ? bias[col] : 0.0f;
#pragma unroll
    for (int r = 0; r < 8; r++) {
        int row = rbase + r;
        float v = acc[r] + bv;
        if (resid) v += resid[row * (size_t)ldc + col];
        C[row * (size_t)ldc + col] = v;
    }
}

// ---------------- causal depthwise conv (K=4) + bias + silu ------------------
__global__ void conv_silu_kernel(const float* __restrict__ xs,  // [T, 2E], u part = cols 0..255
                                 const float* __restrict__ cw,  // [E,1,4]
                                 const float* __restrict__ cb,  // [E]
                                 float* __restrict__ u) {       // [T, E]
    int t = blockIdx.x, e = threadIdx.x;
    float acc = cb[e];
#pragma unroll
    for (int k = 0; k < 4; k++) {
        int tt = t - 3 + k;
        if (tt >= 0) acc += xs[tt * (2 * E_DIM) + e] * cw[e * 4 + k];
    }
    u[t * E_DIM + e] = silu_f(acc);
}

// ---------------- dt_r = u @ w_param[0:8].T  (one wave per (t,r) dot) --------
__global__ void dtr_kernel(const float* __restrict__ u,   // [T,E]
                           const float* __restrict__ wp,  // rows 0..7 of w_param [8,E]
                           float* __restrict__ dtr) {     // [T,8]
    int lane = threadIdx.x & 31;
    int w = blockIdx.x * 8 + (threadIdx.x >> 5);  // 0 .. T*8-1
    int t = w >> 3, r = w & 7;
    const float* up = u + (size_t)t * E_DIM;
    const float* wr = wp + (size_t)r * E_DIM;
    float s = 0.0f;
#pragma unroll
    for (int c = lane; c < E_DIM; c += 32) s += up[c] * wr[c];
#pragma unroll
    for (int m = 16; m > 0; m >>= 1) s += __shfl_xor(s, m, 32);
    if (lane == 0) dtr[w] = s;
}

// ---------------- dt = softplus(dt_r @ w_dt.T + b_dt) / sqrt(E) --------------
__global__ void dt_kernel(const float* __restrict__ dtr,  // [T,8]
                          const float* __restrict__ wdt,  // [E,8]
                          const float* __restrict__ bdt,  // [E]
                          float* __restrict__ dt) {       // [T,E]
    int t = blockIdx.x, e = threadIdx.x;
    float s = bdt[e];
#pragma unroll
    for (int r = 0; r < 8; r++) s += dtr[t * 8 + r] * wdt[e * 8 + r];
    dt[t * E_DIM + e] = softplus_f(s) * (1.0f / 16.0f);  // sqrt(E)=16
}

// ---------------- selective scan + epilogue ----------------------------------
// h[t,e,n] = exp(dt[t,e]) * h[t-1,e,n] + u[t,e]*(exp(dt[t,e])-1) * Bc[t,n]
// y[t,e]   = sum_n h[t,e,n] * Cc[t,n]
// z[t,e]   = (y + u*D[e]) * silu(skip[t,e])     (skip = xs[:,256:])
// One wave32 per channel e; lane holds n = lane + {0,32,64,96}.
// Bc/Cc rows are shared by all 8 waves of a block -> double-buffered async
// staging into LDS with GLOBAL_LOAD_ASYNC_TO_LDS_B128 (ASYNCcnt-tracked).
__global__ void scan_kernel(const float* __restrict__ dt,  // [T,E]
                            const float* __restrict__ u,   // [T,E]
                            const float* __restrict__ Bc,  // [T,H]
                            const float* __restrict__ Cc,  // [T,H]
                            const float* __restrict__ xs,  // [T,2E]
                            const float* __restrict__ Dp,  // [E]
                            float* __restrict__ z) {       // [T,E]
    __shared__ float sB[2][TCH * H_DIM];
    __shared__ float sC[2][TCH * H_DIM];

    int tid = threadIdx.x;
    int lane = tid & 31;
    int e = blockIdx.x * 8 + (tid >> 5);

    float h0 = 0.f, h1 = 0.f, h2 = 0.f, h3 = 0.f;
    float d_e = Dp[e];
    int n0 = lane, n1 = lane + 32, n2 = lane + 64, n3 = lane + 96;

    // issue one chunk (TCH rows of Bc and Cc) as async LDS loads:
    // 256 threads x 16B covers TCH*H_DIM*4 = 4096B per array.
    auto issue = [&](int c) {
        int buf = c & 1;
        unsigned lb = (unsigned)(size_t)&sB[buf][tid * 4];
        unsigned lc = (unsigned)(size_t)&sC[buf][tid * 4];
        const float* gb = Bc + (size_t)c * (TCH * H_DIM) + tid * 4;
        const float* gc = Cc + (size_t)c * (TCH * H_DIM) + tid * 4;
        asm volatile("global_load_async_to_lds_b128 %0, %1, off"
                     :: "v"(lb), "v"(gb) : "memory");
        asm volatile("global_load_async_to_lds_b128 %0, %1, off"
                     :: "v"(lc), "v"(gc) : "memory");
    };

    issue(0);
    const int NCHUNK = T_SEQ / TCH;
    for (int c = 0; c < NCHUNK; c++) {
        if (c + 1 < NCHUNK) {
            issue(c + 1);
            // async loads complete in order: <=2 outstanding => chunk c landed
            asm volatile("s_wait_asynccnt 0x2" ::: "memory");
        } else {
            asm volatile("s_wait_asynccnt 0x0" ::: "memory");
        }
        __syncthreads();
        int buf = c & 1;
        const float* bb = &sB[buf][0];
        const float* cb = &sC[buf][0];
#pragma unroll
        for (int j = 0; j < TCH; j++) {
            int t = c * TCH + j;
            float dte = dt[t * E_DIM + e];
            float ue  = u[t * E_DIM + e];
            float g = __expf(dte);
            float cf = ue * (g - 1.0f);
            const float* bp = bb + j * H_DIM;
            const float* cp = cb + j * H_DIM;
            h0 = fmaf(g, h0, cf * bp[n0]);
            h1 = fmaf(g, h1, cf * bp[n1]);
            h2 = fmaf(g, h2, cf * bp[n2]);
            h3 = fmaf(g, h3, cf * bp[n3]);
            float acc = h0 * cp[n0] + h1 * cp[n1] + h2 * cp[n2] + h3 * cp[n3];
#pragma unroll
            for (int m = 16; m > 0; m >>= 1) acc += __shfl_xor(acc, m, 32);
            if (lane == 0) {
                float skip = xs[t * (2 * E_DIM) + E_DIM + e];
                z[t * E_DIM + e] = (acc + ue * d_e) * silu_f(skip);
            }
        }
        __syncthreads();  // buffer buf gets re-issued at iteration c+1
    }
}

extern "C" void kernel_launch(void* const* d_in, const int* in_sizes, int n_in,
                              void* d_out, int out_size, void* d_ws, size_t ws_size,
                              hipStream_t stream) {
    const float* x_in    = (const float*)d_in[0];   // [1,1024,128]
    const float* w_in    = (const float*)d_in[1];   // [2,512,128]
    const float* b_in    = (const float*)d_in[2];   // [2,512]
    const float* w_out   = (const float*)d_in[3];   // [2,128,256]
    const float* b_out   = (const float*)d_in[4];   // [2,128]
    const float* w_param = (const float*)d_in[5];   // [2,264,256]
    const float* w_dt    = (const float*)d_in[6];   // [2,256,8]
    const float* b_dt    = (const float*)d_in[7];   // [2,256]
    const float* conv_w  = (const float*)d_in[8];   // [2,256,1,4]
    const float* conv_b  = (const float*)d_in[9];   // [2,256]
    const float* Dp      = (const float*)d_in[10];  // [2,256]
    const float* norm_w  = (const float*)d_in[11];  // [2,128]
    const float* fnw     = (const float*)d_in[12];  // [128]
    float* out = (float*)d_out;

    // workspace layout (floats)
    float* ws  = (float*)d_ws;
    float* xn  = ws;                         // 1024*128
    float* xs  = xn  + T_SEQ * H_DIM;        // 1024*512
    float* u   = xs  + T_SEQ * 2 * E_DIM;    // 1024*256
    float* Bc  = u   + T_SEQ * E_DIM;        // 1024*128
    float* Cc  = Bc  + T_SEQ * H_DIM;        // 1024*128
    float* dtr = Cc  + T_SEQ * H_DIM;        // 1024*8
    float* dt  = dtr + T_SEQ * 8;            // 1024*256
    float* z   = dt  + T_SEQ * E_DIM;        // 1024*256
    float* xb0 = z   + T_SEQ * E_DIM;        // 1024*128
    float* xb1 = xb0 + T_SEQ * H_DIM;        // 1024*128
    float* xb[2] = {xb0, xb1};

    const float* xin = x_in;
    for (int i = 0; i < 2; i++) {
        const float* w_in_i  = w_in    + (size_t)i * 512 * 128;
        const float* b_in_i  = b_in    + (size_t)i * 512;
        const float* w_out_i = w_out   + (size_t)i * 128 * 256;
        const float* b_out_i = b_out   + (size_t)i * 128;
        const float* wp_i    = w_param + (size_t)i * 264 * 256;
        const float* w_dt_i  = w_dt    + (size_t)i * 256 * 8;
        const float* b_dt_i  = b_dt    + (size_t)i * 256;
        const float* cw_i    = conv_w  + (size_t)i * 256 * 4;
        const float* cb_i    = conv_b  + (size_t)i * 256;
        const float* D_i     = Dp      + (size_t)i * 256;
        const float* nw_i    = norm_w  + (size_t)i * 128;

        // 1) rmsnorm
        rmsnorm_kernel<<<T_SEQ, H_DIM, 0, stream>>>(xin, nw_i, xn);
        // 2) xs = xn @ w_in.T + b_in  (1024x128 * 128x512)
        wmma_gemm_f32<<<(64 * 32) / 8, 256, 0, stream>>>(
            xn, H_DIM, w_in_i, H_DIM, b_in_i, nullptr, xs, 2 * E_DIM,
            /*tiles_n=*/32, /*K=*/H_DIM);
        // 3) u = silu(conv(xs[:, :256]) + cb)
        conv_silu_kernel<<<T_SEQ, E_DIM, 0, stream>>>(xs, cw_i, cb_i, u);
        // 4) Bc = u @ wp[8:136].T,  Cc = u @ wp[136:264].T  (1024x256 * 256x128)
        wmma_gemm_f32<<<(64 * 8) / 8, 256, 0, stream>>>(
            u, E_DIM, wp_i + 8 * 256, E_DIM, nullptr, nullptr, Bc, H_DIM, 8, E_DIM);
        wmma_gemm_f32<<<(64 * 8) / 8, 256, 0, stream>>>(
            u, E_DIM, wp_i + 136 * 256, E_DIM, nullptr, nullptr, Cc, H_DIM, 8, E_DIM);
        // 5) dt_r = u @ wp[0:8].T   (one wave per output dot)
        dtr_kernel<<<(T_SEQ * 8) / 8, 256, 0, stream>>>(u, wp_i, dtr);
        // 6) dt = softplus(dt_r @ w_dt.T + b_dt) / 16
        dt_kernel<<<T_SEQ, E_DIM, 0, stream>>>(dtr, w_dt_i, b_dt_i, dt);
        // 7) selective scan + gating epilogue -> z (async LDS staged B/C)
        scan_kernel<<<E_DIM / 8, 256, 0, stream>>>(dt, u, Bc, Cc, xs, D_i, z);
        // 8) x_next = z @ w_out.T + b_out + xin  (1024x256 * 256x128, fused residual)
        wmma_gemm_f32<<<(64 * 8) / 8, 256, 0, stream>>>(
            z, E_DIM, w_out_i, E_DIM, b_out_i, xin, xb[i], H_DIM, 8, E_DIM);
        xin = xb[i];
    }
    // final rmsnorm -> d_out
    rmsnorm_kernel<<<T_SEQ, H_DIM, 0, stream>>>(xin, fnw, out);
}